// GCMCGraphConv_84335977824412
// MI455X (gfx1250) — compile-verified
//
#include <hip/hip_runtime.h>
#include <hip/hip_bf16.h>

typedef __attribute__((ext_vector_type(2))) float v2f;
typedef __attribute__((ext_vector_type(8))) float v8f;

#define FEAT 64
#define WAVES_PER_BLOCK 8
#define LDS_STRIDE 68  // 64 + 4 pad: bank-conflict-free b64 reads (banks 4n+2h)

__global__ void gcmc_zero_kernel(float* __restrict__ out, int ntot) {
    int i = blockIdx.x * blockDim.x + threadIdx.x;
    if (i < ntot) out[i] = 0.0f;
}

__global__ void gcmc_scale_kernel(float* __restrict__ out,
                                  const float* __restrict__ ci, int ntot) {
    int i = blockIdx.x * blockDim.x + threadIdx.x;
    if (i < ntot) out[i] *= ci[i >> 6];  // FEAT == 64
}

// One wave processes 16 edges: rf = review_feat[e0:e0+16] @ W^T via
// 4 col-tiles x 16 K-steps of V_WMMA_F32_16X16X4_F32, then fused
// gather + scale + atomic scatter-add.
__global__ void __launch_bounds__(256)
gcmc_edge_kernel(const float* __restrict__ x,
                 const float* __restrict__ weight,
                 const float* __restrict__ w_review,   // [64,64] row-major
                 const float* __restrict__ review_feat,// [E,64]
                 const float* __restrict__ ci,         // [N]
                 const int*   __restrict__ src,
                 const int*   __restrict__ dst,
                 float* __restrict__ out,              // [N,64]
                 int n_edges, int n_groups) {
    __shared__ float ldsW[FEAT * LDS_STRIDE];

    // Stage W_r into padded LDS (coalesced, 16 floats/thread).
    for (int i = threadIdx.x; i < FEAT * FEAT; i += 256) {
        int r = i >> 6, c = i & 63;
        ldsW[r * LDS_STRIDE + c] = w_review[i];
    }
    __syncthreads();

    const int lane = threadIdx.x & 31;
    const int wave = threadIdx.x >> 5;
    const int n    = lane & 15;   // col-within-tile / A-row
    const int h    = lane >> 4;   // lane half selects K pair / M+8

    const int group = blockIdx.x * WAVES_PER_BLOCK + wave;
    if (group >= n_groups) return;
    const int e0 = group * 16;

    // ---- A fragments: lane(n,h) holds A[n][4*kc + 2h + {0,1}] ----
    int arow = e0 + n;
    if (arow >= n_edges) arow = n_edges - 1;  // clamp (tail rows masked below)
    const v2f* __restrict__ ap =
        (const v2f*)(review_feat + (size_t)arow * FEAT + 2 * h);
    v2f afrag[16];
#pragma unroll
    for (int kc = 0; kc < 16; ++kc) afrag[kc] = ap[kc * 2];  // stride 16B

    // ---- per-row edge metadata (row r of D -> edge e0 + r + 8h) ----
    int   sidx[8], didx[8];
    float cs[8];
    bool  valid[8];
#pragma unroll
    for (int r = 0; r < 8; ++r) {
        int e = e0 + r + 8 * h;
        valid[r] = (e < n_edges);
        if (!valid[r]) e = n_edges - 1;
        sidx[r] = src[e];
        didx[r] = dst[e];
        cs[r]   = ci[sidx[r]];
    }

#pragma unroll
    for (int jt = 0; jt < 4; ++jt) {
        const int j0 = jt * 16;
        // B fragment: lane(n,h) needs W[j0+n][4*kc + 2h + {0,1}]
        const float* __restrict__ wl = &ldsW[(j0 + n) * LDS_STRIDE + 2 * h];

        v8f acc = {};
#pragma unroll
        for (int kc = 0; kc < 16; ++kc) {
            v2f b;
            b.x = wl[kc * 4 + 0];
            b.y = wl[kc * 4 + 1];
            acc = __builtin_amdgcn_wmma_f32_16x16x4_f32(
                false, afrag[kc], false, b, (short)0, acc, false, false);
        }

        // ---- epilogue: D[r][lane] -> edge e0+r+8h, feat j0+n ----
        const int feat = j0 + n;
#pragma unroll
        for (int r = 0; r < 8; ++r) {
            if (!valid[r]) continue;
            const size_t so = (size_t)sidx[r] * FEAT + feat;
            float val = (x[so] + weight[so] + acc[r]) * cs[r];
            float* p = out + (size_t)didx[r] * FEAT + feat;
            __hip_atomic_fetch_add(p, val, __ATOMIC_RELAXED,
                                   __HIP_MEMORY_SCOPE_AGENT);
        }
    }
}

extern "C" void kernel_launch(void* const* d_in, const int* in_sizes, int n_in,
                              void* d_out, int out_size, void* d_ws, size_t ws_size,
                              hipStream_t stream) {
    const float* x           = (const float*)d_in[0];
    const float* weight      = (const float*)d_in[1];
    // d_in[2] = w_prob: dead code in the reference aggregation, unused.
    const float* w_review    = (const float*)d_in[3];
    const float* review_feat = (const float*)d_in[4];
    const float* ci          = (const float*)d_in[5];
    const int*   src         = (const int*)d_in[6];
    const int*   dst         = (const int*)d_in[7];
    float*       out         = (float*)d_out;

    const int n_edges = in_sizes[7];               // 1,250,000
    const int ntot    = out_size;                  // 150,000 * 64

    // 1) zero the (poisoned) output accumulator
    gcmc_zero_kernel<<<(ntot + 255) / 256, 256, 0, stream>>>(out, ntot);

    // 2) fused WMMA transform + gather + atomic scatter-sum
    const int n_groups = (n_edges + 15) / 16;      // 78,125 (exact)
    const int n_blocks = (n_groups + WAVES_PER_BLOCK - 1) / WAVES_PER_BLOCK;
    gcmc_edge_kernel<<<n_blocks, 256, 0, stream>>>(
        x, weight, w_review, review_feat, ci, src, dst, out, n_edges, n_groups);

    // 3) final ci[dst] scaling
    gcmc_scale_kernel<<<(ntot + 255) / 256, 256, 0, stream>>>(out, ci, ntot);
}